// ObliviousTreeEmbedding_9801115370131
// MI455X (gfx1250) — compile-verified
//
#include <hip/hip_runtime.h>

// ObliviousTreeEmbedding for MI455X (gfx1250, wave32).
//
// Roofline: output = 8192 x 4096 f32 = 134 MB stores -> ~5.8us @ 23.3 TB/s.
// GEMM is 0.27 GFLOP (negligible) -> use full-precision V_WMMA_F32_16X16X4_F32.
// One fused kernel: WMMA GEMM + bias + entmax(alpha=4, 25 bisection iters)
// -> factors in LDS -> outer-product expansion with coalesced b128 stores.

typedef __attribute__((ext_vector_type(2))) float v2f;
typedef __attribute__((ext_vector_type(4))) float v4f;
typedef __attribute__((ext_vector_type(8))) float v8f;

#define FDIM   512
#define NCOLS  32     // DEPTH * WIDTH
#define OUTW   4096   // 8^4
#define TILE_M 16

// ---- 8-lane group reductions via single-instruction ds_swizzle_b32 --------
// group-of-32 mode: imm = xor_mask<<10 | or_mask<<5 | and_mask.
// xor with 1/2/4 stays inside groups {0-7},{8-15},{16-23},{24-31}.
#define SWZ_XOR(v, imm) \
  __int_as_float(__builtin_amdgcn_ds_swizzle(__float_as_int(v), (imm)))

__device__ __forceinline__ float gsum8(float v) {
  v += SWZ_XOR(v, 0x041F);   // xor 1
  v += SWZ_XOR(v, 0x081F);   // xor 2
  v += SWZ_XOR(v, 0x101F);   // xor 4
  return v;
}
__device__ __forceinline__ float gmax8(float v) {
  v = fmaxf(v, SWZ_XOR(v, 0x041F));
  v = fmaxf(v, SWZ_XOR(v, 0x081F));
  v = fmaxf(v, SWZ_XOR(v, 0x101F));
  return v;
}

// relu(x)^(1/3) via hardware v_log_f32 (log2) / v_exp_f32 (exp2)
__device__ __forceinline__ float cbrt_relu(float x) {
  return x > 0.f
      ? __builtin_amdgcn_exp2f(__builtin_amdgcn_logf(x) * 0.33333333333f)
      : 0.f;
}

// entmax_bisect over an 8-lane group; each lane passes its logit, gets back
// its normalized probability. Faithful to the reference (f_lo fixed, p from
// the final midpoint, ensure_sum_one normalization).
__device__ __forceinline__ float entmax8(float logit) {
  const float z  = 3.0f * logit;          // X * (alpha-1)
  const float mx = gmax8(z);
  float tau  = mx - 1.0f;                 // tau_lo
  const float f_lo = gsum8(cbrt_relu(z - tau)) - 1.0f;
  float dm = 1.0f - 0.001953125f;         // tau_hi - tau_lo = 1 - (1/8)^3
  float p  = 0.0f;
  for (int it = 0; it < 25; ++it) {
    dm *= 0.5f;
    const float tm = tau + dm;
    p = cbrt_relu(z - tm);
    const float fm = gsum8(p) - 1.0f;
    if (fm * f_lo >= 0.0f) tau = tm;
  }
  return p / gsum8(p);
}

__global__ __launch_bounds__(256)
void tree_embed_kernel(const float* __restrict__ X,
                       const float* __restrict__ W,
                       const float* __restrict__ bias,
                       float* __restrict__ out) {
  __shared__ float sF[TILE_M][NCOLS];   // entmax factors, per local row
  __shared__ float sP01[TILE_M][64];    // f0[i0]*f1[i1]

  const int tid  = threadIdx.x;
  const int wave = tid >> 5;
  const int lane = tid & 31;
  const int lh   = lane >> 4;           // half-wave selector
  const int lm   = lane & 15;
  const int rowBase = blockIdx.x * TILE_M;

  // ---------------- Phase 1 (wave 0): WMMA GEMM + entmax -------------------
  if (wave == 0) {
    v8f acc0 = {0,0,0,0,0,0,0,0};       // N = 0..15
    v8f acc1 = {0,0,0,0,0,0,0,0};       // N = 16..31
    // A 16x4 f32 layout: lane -> M=lm, K pair = {2*lh, 2*lh+1}
    const float* xrow = X + (size_t)(rowBase + lm) * FDIM + 2 * lh;
    // B 4x16 layout mirrored: lane -> N=lm, K pair = {2*lh, 2*lh+1}
    const float* w0 = W + (size_t)lm * FDIM + 2 * lh;
    const float* w1 = W + (size_t)(16 + lm) * FDIM + 2 * lh;

    for (int k = 0; k < FDIM; k += 4) {
      const v2f a  = *(const v2f*)(xrow + k);
      const v2f b0 = *(const v2f*)(w0 + k);
      const v2f b1 = *(const v2f*)(w1 + k);
      acc0 = __builtin_amdgcn_wmma_f32_16x16x4_f32(false, a, false, b0,
                                                   (short)0, acc0, false, false);
      acc1 = __builtin_amdgcn_wmma_f32_16x16x4_f32(false, a, false, b1,
                                                   (short)0, acc1, false, false);
    }

    const float bias0 = bias[lm];
    const float bias1 = bias[16 + lm];
    // C/D layout: VGPR r -> M = r (lanes 0-15) or r+8 (lanes 16-31); N = lm.
    for (int r = 0; r < 8; ++r) {
      const float f0 = entmax8(acc0[r] + bias0);
      const float f1 = entmax8(acc1[r] + bias1);
      sF[8 * lh + r][lm]      = f0;
      sF[8 * lh + r][16 + lm] = f1;
    }
  }
  __syncthreads();

  // ---------------- Phase 2 (all threads): P01 table -----------------------
  for (int jj = 0; jj < 4; ++jj) {
    const int e = tid + 256 * jj;       // 1024 entries = 16 rows x 64
    const int r = e >> 6, h = e & 63;
    sP01[r][h] = sF[r][h >> 3] * sF[r][8 + (h & 7)];
  }
  __syncthreads();

  // ---------------- Phase 3 (all 8 waves): expansion + stores --------------
  // Column c = t*128 + 4*lane; c = hi*64 + lo with hi = 2t+lh, lo = 4*lm + j.
  for (int rr = 0; rr < 2; ++rr) {
    const int r = wave * 2 + rr;
    const float  f2  = sF[r][16 + (lm >> 1)];          // i2 = lo>>3 (const in j)
    const float* f3p = &sF[r][24 + 4 * (lm & 1)];      // i3 = 4*(lm&1)+j
    const float g0 = f2 * f3p[0], g1 = f2 * f3p[1];
    const float g2 = f2 * f3p[2], g3 = f2 * f3p[3];
    float* orow = out + (size_t)(rowBase + r) * OUTW + 4 * lane;
    for (int t = 0; t < 32; ++t) {
      const float u = sP01[r][2 * t + lh];
      const v4f val = { u * g0, u * g1, u * g2, u * g3 };
      *(v4f*)(orow + t * 128) = val;                   // global_store_b128
    }
  }
}

extern "C" void kernel_launch(void* const* d_in, const int* in_sizes, int n_in,
                              void* d_out, int out_size, void* d_ws, size_t ws_size,
                              hipStream_t stream) {
  const float* X  = (const float*)d_in[0];   // [8192, 512]
  const float* W  = (const float*)d_in[1];   // [4, 8, 512] -> rows of [32,512]
  const float* bi = (const float*)d_in[2];   // [4, 8] -> flat [32]
  float* out = (float*)d_out;                // [8192, 4096]
  tree_embed_kernel<<<8192 / TILE_M, 256, 0, stream>>>(X, W, bi, out);
}